// H2GCN_23270132809748
// MI455X (gfx1250) — compile-verified
//
#include <hip/hip_runtime.h>

#define N_PTS 30000
#define CH    512
#define KCL   3000
#define STRIDE_ 10

typedef __attribute__((ext_vector_type(2))) float v2f;
typedef __attribute__((ext_vector_type(8))) float v8f;

// ---------------------------------------------------------------- norm(w)
__global__ void norm_kernel(const float* __restrict__ w, float* __restrict__ out_norm) {
    __shared__ float red[256];
    float s = 0.f;
    for (int c = threadIdx.x; c < CH; c += 256) { float v = w[c]; s += v * v; }
    red[threadIdx.x] = s;
    __syncthreads();
    for (int off = 128; off > 0; off >>= 1) {
        if (threadIdx.x < (unsigned)off) red[threadIdx.x] += red[threadIdx.x + off];
        __syncthreads();
    }
    if (threadIdx.x == 0) out_norm[0] = sqrtf(red[0]);
}

// ------------------------------------------- fitness = tanh(x.w / |w|), WMMA f32
// One wave -> 16 rows. A: 16x4 f32 chunk of x (VGPR r <-> k = 2*(lane>=16)+r,
// M = lane&15). B: 4x16 with w chunk replicated across all 16 columns, same
// k-mapping. C/D 16x16 f32: lane group kh holds rows 8*kh + r in acc[r].
__global__ void fitness_wmma_kernel(const float* __restrict__ x,
                                    const float* __restrict__ w,
                                    const float* __restrict__ nrm,
                                    float* __restrict__ fit_ws,
                                    float* __restrict__ fit_out) {
    const int wave = (int)((blockIdx.x * blockDim.x + threadIdx.x) >> 5);
    const int nWaves = N_PTS / 16;              // 1875 exactly
    if (wave >= nWaves) return;                 // wave-uniform: EXEC stays all-1s
    const int lane = threadIdx.x & 31;
    const int m    = lane & 15;
    const int kh   = lane >> 4;                 // 0: k=0,1  1: k=2,3
    const int base = wave * 16;
    const float* xr = x + (size_t)(base + m) * CH + 2 * kh;
    const float* wp = w + 2 * kh;

    v8f acc = {0.f, 0.f, 0.f, 0.f, 0.f, 0.f, 0.f, 0.f};
#pragma unroll 8
    for (int kc = 0; kc < CH; kc += 4) {
        v2f a = *(const v2f*)(xr + kc);
        v2f b = *(const v2f*)(wp + kc);
        acc = __builtin_amdgcn_wmma_f32_16x16x4_f32(false, a, false, b,
                                                    (short)0, acc, false, false);
    }
    // all 16 columns identical -> lanes 0 and 16 publish rows [base+8*kh .. +7]
    if (m == 0) {
        float nv = nrm[0];
#pragma unroll
        for (int r = 0; r < 8; ++r) {
            int row = base + 8 * kh + r;
            float f = tanhf(acc[r] / nv);
            fit_ws[row]  = f;
            fit_out[row] = f;
        }
    }
}

// ---------------------------- stable ascending rank; rank%10==0 -> center k=rank/10
__global__ void rank_kernel(const float* __restrict__ fit,
                            const float* __restrict__ xy,
                            float* __restrict__ centers) {
    __shared__ unsigned long long skey[256];
    const int n = (int)(blockIdx.x * 256 + threadIdx.x);
    unsigned long long mykey = 0ull;
    if (n < N_PTS) {
        unsigned u = __float_as_uint(fit[n]);
        u = (u & 0x80000000u) ? ~u : (u | 0x80000000u);   // order-preserving map
        mykey = ((unsigned long long)u << 32) | (unsigned)n;
    }
    int cnt = 0;
    for (int tb = 0; tb < N_PTS; tb += 256) {
        int idx = tb + (int)threadIdx.x;
        unsigned long long k = ~0ull;                     // never counted as less
        if (idx < N_PTS) {
            unsigned u = __float_as_uint(fit[idx]);
            u = (u & 0x80000000u) ? ~u : (u | 0x80000000u);
            k = ((unsigned long long)u << 32) | (unsigned)idx;
        }
        skey[threadIdx.x] = k;
        __syncthreads();
        if (n < N_PTS) {
#pragma unroll 8
            for (int j = 0; j < 256; ++j) cnt += (skey[j] < mykey) ? 1 : 0;
        }
        __syncthreads();
    }
    if (n < N_PTS && (cnt % STRIDE_) == 0) {
        int k = cnt / STRIDE_;                            // 0..2999
        centers[3 * k + 0] = xy[2 * n + 0];
        centers[3 * k + 1] = xy[2 * n + 1];
        centers[3 * k + 2] = fit[n];
    }
}

// -------------------- cluster[n] = argmin_k sqrt(dx^2+dy^2) + |df| (first-min wins)
__global__ void assign_kernel(const float* __restrict__ centers,
                              const float* __restrict__ xy,
                              const float* __restrict__ fit,
                              int* __restrict__ cl_ws,
                              float* __restrict__ cl_out) {
    __shared__ float c[3 * KCL];                          // 36 KB, LDS-resident
    for (int i = threadIdx.x; i < 3 * KCL; i += 256) c[i] = centers[i];
    __syncthreads();
    const int n = (int)(blockIdx.x * 256 + threadIdx.x);
    if (n >= N_PTS) return;
    float px = xy[2 * n], py = xy[2 * n + 1], pf = fit[n];
    float best = 3.4e38f; int bi = 0;
    for (int k = 0; k < KCL; ++k) {                       // uniform k: LDS broadcast
        float dx = c[3 * k + 0] - px;
        float dy = c[3 * k + 1] - py;
        float d  = sqrtf(dx * dx + dy * dy) + fabsf(c[3 * k + 2] - pf);
        if (d < best) { best = d; bi = k; }
    }
    cl_ws[n]  = bi;
    cl_out[n] = (float)bi;
}

// -------- deterministic segment mean: one block per cluster, ballot-ordered members
__global__ void segmean_kernel(const float* __restrict__ x,
                               const float* __restrict__ xy,
                               const int* __restrict__ cl,
                               float* __restrict__ out_x,
                               float* __restrict__ out_xy) {
    __shared__ int wcnt[8];
    __shared__ int mlist[256];
    const int k    = (int)blockIdx.x;
    const int tid  = (int)threadIdx.x;
    const int lane = tid & 31;
    const int wid  = tid >> 5;
    float acc0 = 0.f, acc1 = 0.f, sxy = 0.f;
    int cnt = 0;
    for (int nb = 0; nb < N_PTS; nb += 256) {
        int idx = nb + tid;
        bool match = (idx < N_PTS) && (cl[idx] == k);
        unsigned mb = __builtin_amdgcn_ballot_w32(match); // wave32 ballot
        if (lane == 0) wcnt[wid] = __popc(mb);
        __syncthreads();
        int before = __popc(mb & (lane ? (0xFFFFFFFFu >> (32 - lane)) : 0u));
        int total = 0;
#pragma unroll
        for (int wv = 0; wv < 8; ++wv) {
            if (wv < wid) before += wcnt[wv];
            total += wcnt[wv];
        }
        if (match) mlist[before] = idx;                   // ascending-n order
        __syncthreads();
        for (int j = 0; j < total; ++j) {                 // fixed order => deterministic
            int m = mlist[j];
            const float* xr = x + (size_t)m * CH;
            acc0 += xr[tid];
            acc1 += xr[tid + 256];
            if (tid < 2) sxy += xy[2 * m + tid];
        }
        cnt += total;
        __syncthreads();
    }
    float denom = (cnt > 0) ? (float)cnt : 1.f;
    out_x[(size_t)k * CH + tid]       = acc0 / denom;
    out_x[(size_t)k * CH + tid + 256] = acc1 / denom;
    if (tid < 2) out_xy[2 * k + tid] = sxy / denom;
}

extern "C" void kernel_launch(void* const* d_in, const int* in_sizes, int n_in,
                              void* d_out, int out_size, void* d_ws, size_t ws_size,
                              hipStream_t stream) {
    const float* x  = (const float*)d_in[0];   // (N, C)
    const float* xy = (const float*)d_in[1];   // (N, 2)
    const float* w  = (const float*)d_in[2];   // (1, C)

    float* out    = (float*)d_out;             // concat: new_x, new_xy, fitness, cluster
    float* out_x  = out;                       // K*C
    float* out_xy = out + (size_t)KCL * CH;    // 2K
    float* out_f  = out_xy + 2 * KCL;          // N
    float* out_c  = out_f + N_PTS;             // N (cluster ids as floats)

    float* wsf   = (float*)d_ws;               // fitness   [N]
    float* wsn   = wsf + N_PTS;                // norm      [1] (+pad)
    float* wsctr = wsn + 4;                    // centers   [3K]
    int*   wscl  = (int*)(wsctr + 3 * KCL);    // cluster   [N]

    norm_kernel<<<1, 256, 0, stream>>>(w, wsn);
    fitness_wmma_kernel<<<235, 256, 0, stream>>>(x, w, wsn, wsf, out_f); // 1880 waves, 1875 active
    rank_kernel<<<118, 256, 0, stream>>>(wsf, xy, wsctr);
    assign_kernel<<<118, 256, 0, stream>>>(wsctr, xy, wsf, wscl, out_c);
    segmean_kernel<<<KCL, 256, 0, stream>>>(x, xy, wscl, out_x, out_xy);
}